// SSTInputLayer_20976620273933
// MI455X (gfx1250) — compile-verified
//
#include <hip/hip_runtime.h>
#include <stdint.h>

// ---- static config (mirrors reference) ----
#define WIN_XY              12
#define SHIFT_XY            6
#define MAX_WIN_Y_          40
#define MAX_WIN_PER_SAMPLE_ 1600
#define NWIN_BINS           6400   // BATCH(4) * MAX_WIN_PER_SAMPLE(1600)
#define NB                  6401   // + sentinel bin for masked-out voxels
#define SENT_BIN            6400
#define CHUNK               4096
#define NTHREADS            256
#define TDM_ROWS            64     // rows per TDM tile: 64 * 512B = 32KB

// TDM builtin availability (device pass only; host pass uses fallback body)
#if defined(__AMDGCN__) && defined(__has_builtin)
# if __has_builtin(__builtin_amdgcn_tensor_load_to_lds)
#  define HAVE_TDM 1
# endif
#endif
#ifndef HAVE_TDM
# define HAVE_TDM 0
#endif

// Workspace requirement: 7*n + 3*NB + numChunks*NB ints  (~10.5 MB for n=300000)

// ---------------------------------------------------------------- windows
__global__ void k_windows(const int* __restrict__ coors, int n,
                          int* __restrict__ win0, int* __restrict__ win1,
                          float* __restrict__ o_bwi0, float* __restrict__ o_bwi1) {
  int i = blockIdx.x * blockDim.x + threadIdx.x;
  if (i >= n) return;
  int b = coors[4 * i + 0];
  int y = coors[4 * i + 2];
  int x = coors[4 * i + 3];
  int w0 = b * MAX_WIN_PER_SAMPLE_ + (x / WIN_XY) * MAX_WIN_Y_ + (y / WIN_XY);
  int x1 = x + SHIFT_XY, y1 = y + SHIFT_XY;
  int w1 = b * MAX_WIN_PER_SAMPLE_ + (x1 / WIN_XY) * MAX_WIN_Y_ + (y1 / WIN_XY);
  win0[i] = w0; win1[i] = w1;
  o_bwi0[i] = (float)w0; o_bwi1[i] = (float)w1;
}

// validity predicate: mode 0 = all, 1 = maskA!=0, 2 = maskA!=0 && lvlA==dl
__device__ __forceinline__ bool is_valid(int mode, const int* maskA, const int* lvlA,
                                         int dl, int i) {
  if (mode == 0) return true;
  if (!maskA[i]) return false;
  if (mode == 1) return true;
  return lvlA[i] == dl;
}

// ------------------------------------------------- ranker pass 1: chunk histogram
__global__ void k_hist(const int* __restrict__ win, const int* maskA, const int* lvlA,
                       int mode, int dl, int n, int* __restrict__ chunkHist) {
  __shared__ int cnt[NB];                       // 25.6 KB of the 320 KB WGP LDS
  for (int b = threadIdx.x; b < NB; b += blockDim.x) cnt[b] = 0;
  __syncthreads();
  int base = blockIdx.x * CHUNK;
  for (int k = threadIdx.x; k < CHUNK; k += blockDim.x) {
    int i = base + k;
    if (i < n) {
      int w = is_valid(mode, maskA, lvlA, dl, i) ? win[i] : SENT_BIN;
      atomicAdd(&cnt[w], 1);
    }
  }
  __syncthreads();
  int* row = chunkHist + (size_t)blockIdx.x * NB;
  for (int b = threadIdx.x; b < NB; b += blockDim.x) row[b] = cnt[b];
}

// ------------------------------------- ranker pass 2: exclusive scan over chunks
__global__ void k_scan(int* __restrict__ chunkHist, int numChunks,
                       int* __restrict__ counts) {
  int b = blockIdx.x * blockDim.x + threadIdx.x;
  if (b >= NB) return;
  int run = 0;
  for (int ch = 0; ch < numChunks; ++ch) {
    size_t idx = (size_t)ch * NB + b;
    int t = chunkHist[idx];
    chunkHist[idx] = run;
    run += t;
  }
  counts[b] = run;
}

// ------------------- ranker pass 3: deterministic inner rank (counting-sort rank)
__global__ void k_rank(const int* __restrict__ win, const int* maskA, const int* lvlA,
                       int mode, int dl, int n, const int* __restrict__ chunkHist,
                       int* __restrict__ inner) {
  __shared__ int s_win[CHUNK];                  // 16 KB
  __shared__ int s_cnt[NB];                     // 25.6 KB
  int base = blockIdx.x * CHUNK;
  for (int b = threadIdx.x; b < NB; b += blockDim.x)
    s_cnt[b] = chunkHist[(size_t)blockIdx.x * NB + b];
  for (int k = threadIdx.x; k < CHUNK; k += blockDim.x) {
    int i = base + k;
    int w = SENT_BIN;
    if (i < n && is_valid(mode, maskA, lvlA, dl, i)) w = win[i];
    s_win[k] = w;
  }
  __syncthreads();
  // 16 ordered segments of 256 elements: original-order stable rank
  for (int seg = 0; seg < CHUNK / NTHREADS; ++seg) {
    int e = seg * NTHREADS + threadIdx.x;
    int w = s_win[e];
    int local = 0;
    for (int j = seg * NTHREADS; j < e; ++j) local += (s_win[j] == w) ? 1 : 0;
    int b0 = s_cnt[w];                          // reads before any add this segment
    __syncthreads();
    atomicAdd(&s_cnt[w], 1);                    // order-independent counter advance
    __syncthreads();
    int i = base + e;
    if (i < n) inner[i] = b0 + local;
  }
}

// --------------------------------------------------------------- drop decision
__global__ void k_drop(const int* __restrict__ win, const int* __restrict__ inner,
                       const int* __restrict__ counts, const int* maskA, int mode,
                       int n, int* __restrict__ keep_s, int* __restrict__ lvl_s,
                       float* __restrict__ o_lvl) {
  int i = blockIdx.x * blockDim.x + threadIdx.x;
  if (i >= n) return;
  bool valid = (mode == 0) || (maskA[i] != 0);
  int lvl = -1, target = 0, kp = 0;
  if (valid) {
    int cw = counts[win[i]];
    if (cw < 30)          { lvl = 0; target = 30;  }
    else if (cw < 60)     { lvl = 1; target = 60;  }
    else if (cw < 100000) { lvl = 2; target = 100; }
    kp = (inner[i] < target) ? 1 : 0;
  }
  keep_s[i] = kp;
  lvl_s[i] = lvl;
  o_lvl[i] = (float)lvl;
}

__global__ void k_combine(const int* __restrict__ keep0, int* __restrict__ keepC,
                          float* __restrict__ o_keep, int n) {
  int i = blockIdx.x * blockDim.x + threadIdx.x;
  if (i >= n) return;
  int k = keep0[i] & keepC[i];
  keepC[i] = k;
  o_keep[i] = (float)k;
}

__global__ void k_zero_i(int* __restrict__ p, int n) {
  int i = blockIdx.x * blockDim.x + threadIdx.x;
  if (i < n) p[i] = 0;
}
__global__ void k_fill_f(float* __restrict__ p, float v, int n) {
  int i = blockIdx.x * blockDim.x + threadIdx.x;
  if (i < n) p[i] = v;
}

// ----------------------------------------------- dense window rank (make_continuous)
__global__ void k_presence(const int* __restrict__ win, const int* __restrict__ keepC,
                           const int* __restrict__ lvlA, int dl, int n,
                           int* __restrict__ pres) {
  int i = blockIdx.x * blockDim.x + threadIdx.x;
  if (i >= n) return;
  if (keepC[i] && lvlA[i] == dl) pres[win[i]] = 1;   // idempotent stores
}

__global__ void k_prefix(const int* __restrict__ pres, int* __restrict__ conti) {
  __shared__ int part[NTHREADS];
  const int per = (NWIN_BINS + NTHREADS - 1) / NTHREADS;   // 25 bins / thread
  int t = threadIdx.x;
  int s = 0;
  for (int k = 0; k < per; ++k) { int b = t * per + k; if (b < NWIN_BINS) s += pres[b]; }
  part[t] = s;
  __syncthreads();
  if (t == 0) {
    int r = 0;
    for (int k = 0; k < NTHREADS; ++k) { int v = part[k]; part[k] = r; r += v; }
  }
  __syncthreads();
  int run = part[t];
  for (int k = 0; k < per; ++k) {
    int b = t * per + k;
    if (b < NWIN_BINS) { int v = pres[b]; conti[b] = run; run += v; }
  }
}

__global__ void k_f2w(const int* __restrict__ win, const int* __restrict__ keepC,
                      const int* __restrict__ lvlA, int dl, int max_tok,
                      const int* __restrict__ conti, const int* __restrict__ inner,
                      int n, float* __restrict__ o_f2w) {
  int i = blockIdx.x * blockDim.x + threadIdx.x;
  if (i >= n) return;
  if (keepC[i] && lvlA[i] == dl)
    o_f2w[i] = (float)(conti[win[i]] * max_tok + inner[i]);
}

// ============================ feature streaming (bandwidth-dominant, ~307 MB) ====

// -- path A: per-wave async global->LDS (known-good on this toolchain) -----------
__global__ void __launch_bounds__(NTHREADS)
k_feat_async(const float* __restrict__ feat, const int* __restrict__ keepC,
             int n, float* __restrict__ out) {
  __shared__ float4 stage[8][2][32];            // 8 KB: 8 waves x 2 bufs x 512 B
  const int lane = threadIdx.x & 31;
  const int wave = threadIdx.x >> 5;
  unsigned ldsA = (unsigned)(size_t)&stage[wave][0][lane];
  unsigned ldsB = (unsigned)(size_t)&stage[wave][1][lane];
  long long stride = (long long)gridDim.x * 8;
  long long row = (long long)blockIdx.x * 8 + wave;
  if (row >= n) return;
  {
    unsigned long long g =
        (unsigned long long)(size_t)((const char*)(feat + row * 128) + lane * 16);
    asm volatile("global_load_async_to_lds_b128 %0, %1, off"
                 :: "v"(ldsA), "v"(g) : "memory");
  }
  int buf = 0;
  for (;;) {
    long long nrow = row + stride;
    if (nrow < n) {
      unsigned nxt = buf ? ldsA : ldsB;
      unsigned long long g =
          (unsigned long long)(size_t)((const char*)(feat + nrow * 128) + lane * 16);
      asm volatile("global_load_async_to_lds_b128 %0, %1, off"
                   :: "v"(nxt), "v"(g) : "memory");
      asm volatile("s_wait_asynccnt 0x1" ::: "memory");   // current buf is ready
    } else {
      asm volatile("s_wait_asynccnt 0x0" ::: "memory");
    }
    float4 v = stage[wave][buf][lane];
    float4 z = make_float4(0.f, 0.f, 0.f, 0.f);
    float4 o = keepC[row] ? v : z;
    *((float4*)(out + row * 128) + lane) = o;
    if (nrow >= n) break;
    row = nrow;
    buf ^= 1;
  }
}

// -- path B: Tensor Data Mover bulk tiles (1 instruction per 32 KB 2D tile) ------
#if HAVE_TDM
typedef unsigned int v4u __attribute__((ext_vector_type(4)));
typedef int v8i __attribute__((ext_vector_type(8)));
typedef int v4i __attribute__((ext_vector_type(4)));

// D# per CDNA5 ISA: 2D tensor of 4-byte elements,
// row length 128 elements (512 B), tile = 128 x rows.
__device__ __forceinline__ void tdm_issue(unsigned lds_off, const float* gsrc,
                                          unsigned rows, unsigned tensor_rows) {
  unsigned long long ga = (unsigned long long)(size_t)gsrc;
  v4u g0;
  g0.x = 1u;                                           // count=1, user mode
  g0.y = lds_off;                                      // lds_addr (bytes)
  g0.z = (unsigned)(ga & 0xffffffffu);                 // global_addr[31:0]
  g0.w = (unsigned)((ga >> 32) & 0x01ffffffu) | (2u << 30); // addr[56:32]|type=2
  v8i g1;
  g1[0] = (int)(2u << 16);                             // data_size=4B
  g1[1] = (int)(128u << 16);                           // tensor_dim0[15:0]=128
  g1[2] = (int)((tensor_rows & 0xffffu) << 16);        // dim0 hi=0 | tensor_dim1 lo
  g1[3] = (int)((tensor_rows >> 16) | (128u << 16));   // tensor_dim1 hi | tile_dim0=128
  g1[4] = (int)(rows & 0xffffu);                       // tile_dim1=rows, tile_dim2=0
  g1[5] = 128;                                         // tensor_dim0_stride[31:0]=128
  g1[6] = 0;                                           // stride0 hi | stride1 lo
  g1[7] = 0;                                           // stride1 hi
  v4i gz = {0, 0, 0, 0};
#if __clang_major__ >= 23
  v8i gz8 = {0, 0, 0, 0, 0, 0, 0, 0};
  __builtin_amdgcn_tensor_load_to_lds(g0, g1, gz, gz, gz8, 0);
#else
  __builtin_amdgcn_tensor_load_to_lds(g0, g1, gz, gz, 0);
#endif
}
#endif // HAVE_TDM

__global__ void __launch_bounds__(NTHREADS)
k_feat_tdm(const float* __restrict__ feat, const int* __restrict__ keepC,
           int n, float* __restrict__ out) {
#if HAVE_TDM
  __shared__ float4 stage[2][TDM_ROWS * 32];           // 2 x 32 KB double buffer
  const int numTiles = (n + TDM_ROWS - 1) / TDM_ROWS;
  unsigned lds0 = (unsigned)(size_t)&stage[0][0];
  unsigned lds1 = (unsigned)(size_t)&stage[1][0];
  int t0 = blockIdx.x;
  if (t0 >= numTiles) return;                          // uniform per block
  const int stride = gridDim.x;
  // prologue: wave 0 issues DMA for first tile (TDM ignores EXEC -> branch-gate
  // per wave so exactly one DMA is issued per block)
  if ((threadIdx.x >> 5) == 0) {
    int r0 = t0 * TDM_ROWS;
    int rows = n - r0; if (rows > TDM_ROWS) rows = TDM_ROWS;
    tdm_issue(lds0, feat + (size_t)r0 * 128, (unsigned)rows, (unsigned)(n - r0));
  }
  int buf = 0;
  for (int t = t0; t < numTiles; t += stride) {
    int tn = t + stride;
    if ((threadIdx.x >> 5) == 0) {
      if (tn < numTiles) {                             // prefetch next tile
        int rn = tn * TDM_ROWS;
        int rows = n - rn; if (rows > TDM_ROWS) rows = TDM_ROWS;
        tdm_issue(buf ? lds0 : lds1, feat + (size_t)rn * 128,
                  (unsigned)rows, (unsigned)(n - rn));
        __builtin_amdgcn_s_wait_tensorcnt(1);          // oldest tile landed
      } else {
        __builtin_amdgcn_s_wait_tensorcnt(0);
      }
    }
    __syncthreads();                                   // publish LDS to all waves
    int r0 = t * TDM_ROWS;
    int rows = n - r0; if (rows > TDM_ROWS) rows = TDM_ROWS;
    for (int k = threadIdx.x; k < rows * 32; k += NTHREADS) {
      int r = r0 + (k >> 5);
      float4 v = stage[buf][k];
      float4 z = make_float4(0.f, 0.f, 0.f, 0.f);
      *((float4*)(out + (size_t)r * 128) + (k & 31)) = keepC[r] ? v : z;
    }
    __syncthreads();                                   // buffer free for reuse
    buf ^= 1;
  }
#else
  // fallback body (also what the host pass parses): vectorized masked copy
  long long total4 = (long long)n * 32;
  long long k = (long long)blockIdx.x * blockDim.x + threadIdx.x;
  long long gstride = (long long)gridDim.x * blockDim.x;
  for (; k < total4; k += gstride) {
    long long r = k >> 5;
    float4 v = ((const float4*)feat)[k];
    float4 z = make_float4(0.f, 0.f, 0.f, 0.f);
    ((float4*)out)[k] = keepC[r] ? v : z;
  }
#endif
}

// generic fallback (c != 128)
__global__ void k_feat_scalar(const float* __restrict__ feat,
                              const int* __restrict__ keepC,
                              long long total, int c, float* __restrict__ out) {
  long long i = (long long)blockIdx.x * blockDim.x + threadIdx.x;
  long long stride = (long long)gridDim.x * blockDim.x;
  for (; i < total; i += stride) out[i] = keepC[i / c] ? feat[i] : 0.f;
}

// ---------------------------------------------------------------------- launch
extern "C" void kernel_launch(void* const* d_in, const int* in_sizes, int n_in,
                              void* d_out, int out_size, void* d_ws, size_t ws_size,
                              hipStream_t stream) {
  (void)n_in; (void)out_size; (void)ws_size;
  const float* feat = (const float*)d_in[0];
  const int* coors  = (const int*)d_in[1];
  const int n = in_sizes[1] / 4;
  const int c = (n > 0) ? in_sizes[0] / n : 0;
  if (n <= 0) return;

  float* o_feat = (float*)d_out;
  float* o_keep = o_feat + (size_t)n * c;
  float* o_lvl0 = o_keep + n;
  float* o_lvl1 = o_lvl0 + n;
  float* o_bwi0 = o_lvl1 + n;
  float* o_bwi1 = o_bwi0 + n;
  float* o_f2w0 = o_bwi1 + n;
  float* o_f2w1 = o_f2w0 + n;

  int* wsp = (int*)d_ws;
  int* win0   = wsp; wsp += n;
  int* win1   = wsp; wsp += n;
  int* inner  = wsp; wsp += n;
  int* keep0  = wsp; wsp += n;
  int* keepC  = wsp; wsp += n;
  int* lvl0   = wsp; wsp += n;
  int* lvl1   = wsp; wsp += n;
  int* counts = wsp; wsp += NB;
  int* pres   = wsp; wsp += NB;
  int* conti  = wsp; wsp += NB;
  int* chunkHist = wsp;

  const int numChunks = (n + CHUNK - 1) / CHUNK;
  const int gN    = (n + NTHREADS - 1) / NTHREADS;
  const int gScan = (NB + NTHREADS - 1) / NTHREADS;

  k_windows<<<gN, NTHREADS, 0, stream>>>(coors, n, win0, win1, o_bwi0, o_bwi1);

  // shift 0 stats over all voxels
  k_hist<<<numChunks, NTHREADS, 0, stream>>>(win0, nullptr, nullptr, 0, 0, n, chunkHist);
  k_scan<<<gScan, NTHREADS, 0, stream>>>(chunkHist, numChunks, counts);
  k_rank<<<numChunks, NTHREADS, 0, stream>>>(win0, nullptr, nullptr, 0, 0, n, chunkHist, inner);
  k_drop<<<gN, NTHREADS, 0, stream>>>(win0, inner, counts, nullptr, 0, n, keep0, lvl0, o_lvl0);

  // shift 1 stats over shift-0 survivors
  k_hist<<<numChunks, NTHREADS, 0, stream>>>(win1, keep0, nullptr, 1, 0, n, chunkHist);
  k_scan<<<gScan, NTHREADS, 0, stream>>>(chunkHist, numChunks, counts);
  k_rank<<<numChunks, NTHREADS, 0, stream>>>(win1, keep0, nullptr, 1, 0, n, chunkHist, inner);
  k_drop<<<gN, NTHREADS, 0, stream>>>(win1, inner, counts, keep0, 1, n, keepC, lvl1, o_lvl1);

  k_combine<<<gN, NTHREADS, 0, stream>>>(keep0, keepC, o_keep, n);

  k_fill_f<<<gN, NTHREADS, 0, stream>>>(o_f2w0, -1.0f, n);
  k_fill_f<<<gN, NTHREADS, 0, stream>>>(o_f2w1, -1.0f, n);

  const int max_tok_tab[3] = {30, 60, 100};
  for (int s = 0; s < 2; ++s) {
    const int* wins = s ? win1 : win0;
    int* lvls       = s ? lvl1 : lvl0;
    float* o_f2w    = s ? o_f2w1 : o_f2w0;
    for (int dl = 0; dl < 3; ++dl) {
      k_zero_i<<<gScan, NTHREADS, 0, stream>>>(pres, NB);
      k_presence<<<gN, NTHREADS, 0, stream>>>(wins, keepC, lvls, dl, n, pres);
      k_prefix<<<1, NTHREADS, 0, stream>>>(pres, conti);
      k_hist<<<numChunks, NTHREADS, 0, stream>>>(wins, keepC, lvls, 2, dl, n, chunkHist);
      k_scan<<<gScan, NTHREADS, 0, stream>>>(chunkHist, numChunks, counts);
      k_rank<<<numChunks, NTHREADS, 0, stream>>>(wins, keepC, lvls, 2, dl, n, chunkHist, inner);
      k_f2w<<<gN, NTHREADS, 0, stream>>>(wins, keepC, lvls, dl, max_tok_tab[dl],
                                         conti, inner, n, o_f2w);
    }
  }

  if (c == 128) {
    int numTiles = (n + TDM_ROWS - 1) / TDM_ROWS;
    int blocks = numTiles < 2048 ? numTiles : 2048;
    k_feat_tdm<<<blocks, NTHREADS, 0, stream>>>(feat, keepC, n, o_feat);
  } else {
    long long total = (long long)n * c;
    k_feat_scalar<<<2048, NTHREADS, 0, stream>>>(feat, keepC, total, c, o_feat);
  }
}